// HR2HK_20538533609907
// MI455X (gfx1250) — compile-verified
//
#include <hip/hip_runtime.h>
#include <stdint.h>
#include <stddef.h>

// ---------------------------------------------------------------------------
// HR2HK on gfx1250: build H(k) [K, N*13, N*13] complex64 (interleaved floats).
// Bandwidth-bound: dominant cost is the 354MB zero-fill of the dense output.
// Hermitianization is fused into the scatter (source entry -> (i,j) block with
// phase, (j,i) block with conj(phase)); no dense re-read.
// CDNA5 paths: v_wmma_f32_16x16x4_f32 for the k.R phase GEMM, TDM
// tensor_load_to_lds for staging edge feature tiles, hardware f32 atomics.
// ---------------------------------------------------------------------------

#define FEAT_DIM 107
#define NORB 13
#define EPB 16   // edges per block in the scatter kernel

typedef float v2f __attribute__((ext_vector_type(2)));
typedef float v4f __attribute__((ext_vector_type(4)));
typedef float v8f __attribute__((ext_vector_type(8)));
typedef unsigned int u32x4 __attribute__((ext_vector_type(4)));
typedef int i32x4 __attribute__((ext_vector_type(4)));
typedef int i32x8 __attribute__((ext_vector_type(8)));

#if __has_builtin(__builtin_amdgcn_wmma_f32_16x16x4_f32)
#define HAVE_WMMA_F32X4 1
#endif
#if __has_builtin(__builtin_amdgcn_tensor_load_to_lds)
#define HAVE_TDM 1
#endif

// Orbital structure: l = [0,0,1,1,2], dims = [1,1,3,3,5], NORB = 13.
__constant__ int c_OFFS[5] = {0, 1, 2, 5, 8};
__constant__ int c_DIMS[5] = {1, 1, 3, 3, 5};
// Flat feature offset of upper-tri orbital pair (i,j); -1 for lower triangle.
__constant__ int c_PAIR_OFF[25] = {
     0,  1,  2,  5,  8,
    -1, 13, 14, 17, 20,
    -1, -1, 25, 34, 43,
    -1, -1, -1, 58, 67,
    -1, -1, -1, -1, 82};

__device__ __forceinline__ int orb_of(int p) {
    return (p < 1) ? 0 : (p < 2) ? 1 : (p < 5) ? 2 : (p < 8) ? 3 : 4;
}
__device__ __forceinline__ int feat_idx(int i, int j, int p, int q) {
    return c_PAIR_OFF[i * 5 + j] + (p - c_OFFS[i]) * c_DIMS[j] + (q - c_OFFS[j]);
}

// ---------------------------------------------------------------------------
// 1) Streaming zero-fill of the dense output (non-temporal b128 stores).
// ---------------------------------------------------------------------------
__global__ __launch_bounds__(256) void zero_fill_kernel(float* __restrict__ out, long n) {
    long n4 = n >> 2;
    long i = (long)blockIdx.x * blockDim.x + threadIdx.x;
    long stride = (long)gridDim.x * blockDim.x;
    v4f z = {0.f, 0.f, 0.f, 0.f};
    v4f* o4 = (v4f*)out;
    for (long t = i; t < n4; t += stride)
        __builtin_nontemporal_store(z, o4 + t);
    if (i == 0)
        for (long t = n4 << 2; t < n; ++t) out[t] = 0.f;
}

// ---------------------------------------------------------------------------
// 2) phase[k,e] = exp(-2*pi*i * dot(kpoints[k], shift[e])), via WMMA f32 16x16x4.
//    A (16x4): rows = k-points (padded with zeros), cols = xyz (+0 pad).
//    B (4x16): rows = xyz (+0 pad), cols = 16 edges of this tile.
//    One wave (32 lanes) per 16 edges.
// ---------------------------------------------------------------------------
__global__ __launch_bounds__(32) void phase_wmma_kernel(
        const float* __restrict__ kpts,   // [K,3]
        const float* __restrict__ shifts, // [E,3]
        float* __restrict__ phase,        // [K,E,2] interleaved re/im
        int E, int K) {
    const int lane = threadIdx.x;
    const int m = lane & 15;
    const bool hi = lane >= 16;
    const int e0 = blockIdx.x * 16;

    // A-matrix operand: lanes 0-15 -> {A[m][0], A[m][1]}, lanes 16-31 -> {A[m][2], A[m][3]=0}
    int mk = (m < K) ? m : 0;
    float k0 = kpts[mk * 3 + 0], k1 = kpts[mk * 3 + 1], k2 = kpts[mk * 3 + 2];
    float mval = (m < K) ? 1.f : 0.f;
    v2f a;
    a.x = hi ? (k2 * mval) : (k0 * mval);
    a.y = hi ? 0.f : (k1 * mval);

    // B-matrix operand: col n = lane%16 -> edge e0+n
    int e = e0 + m;
    int ec = (e < E) ? e : 0;
    float s0 = shifts[ec * 3 + 0], s1 = shifts[ec * 3 + 1], s2 = shifts[ec * 3 + 2];
    v2f b;
    b.x = hi ? s2 : s0;
    b.y = hi ? 0.f : s1;

    v8f d;
#if defined(HAVE_WMMA_F32X4)
    v8f c = {};
    // D[m][n] = dot(kpoint_m, shift_n); lanes 0-15 hold rows 0-7 in d[0..7]
    d = __builtin_amdgcn_wmma_f32_16x16x4_f32(false, a, false, b,
                                              (short)0, c, false, false);
#else
    for (int r = 0; r < 8; ++r) d[r] = 0.f;
    if (!hi) {
        for (int r = 0; r < K; ++r)
            d[r] = kpts[r * 3 + 0] * s0 + kpts[r * 3 + 1] * s1 + kpts[r * 3 + 2] * s2;
    }
#endif

    if (!hi && e < E) {
        const float TWO_PI = 6.28318530717958647692f;
        for (int k = 0; k < K && k < 8; ++k) {
            float t = TWO_PI * d[k];
            float s, cc;
            __sincosf(t, &s, &cc);
            size_t idx = 2 * ((size_t)k * E + e);
            phase[idx + 0] = cc;    // cos(2*pi*x)
            phase[idx + 1] = -s;    // exp(-i*2*pi*x) -> imag = -sin
        }
    }
}

// ---------------------------------------------------------------------------
// 3) Onsite diagonal blocks: out[k][a-block diag](p,q) = B[p][q] + B[q][p]
//    (real; hermitianization of the real onsite block). Plain stores: runs
//    after zero-fill, before the atomic edge scatter (stream-ordered).
// ---------------------------------------------------------------------------
__global__ __launch_bounds__(256) void onsite_kernel(
        const float* __restrict__ ons,  // [N,FEAT_DIM]
        float* __restrict__ out, int N, int K, int M) {
    int t = blockIdx.x * blockDim.x + threadIdx.x;
    int total = N * NORB * NORB;
    if (t >= total) return;
    int a = t / (NORB * NORB);
    int pe = t - a * (NORB * NORB);
    int p = pe / NORB, q = pe - p * NORB;
    int i = orb_of(p), j = orb_of(q);
    float val = 0.f;
    const float* f = ons + (size_t)a * FEAT_DIM;
    if (i <= j) {
        float s = (i == j) ? 0.5f : 1.0f;
        val += s * f[feat_idx(i, j, p, q)];
    }
    if (j <= i) {   // transpose term B[q][p]
        float s = (i == j) ? 0.5f : 1.0f;
        val += s * f[feat_idx(j, i, q, p)];
    }
    size_t row = (size_t)a * NORB + p, col = (size_t)a * NORB + q;
    size_t mm = (size_t)M * M;
    for (int k = 0; k < K; ++k)
        out[2 * ((size_t)k * mm + row * M + col)] = val;  // imag stays 0
}

// ---------------------------------------------------------------------------
// 4) Edge scatter: stage 16 edges' feature rows into LDS via TDM, then each
//    nonzero assembled entry h(p,q) contributes
//      out[k][(ai*13+p, aj*13+q)] +=  phase * h
//      out[k][(aj*13+q, ai*13+p)] += conj(phase) * h
//    using hardware global_atomic_add_f32 (edges may collide).
// ---------------------------------------------------------------------------
__global__ __launch_bounds__(256) void edge_scatter_kernel(
        const float* __restrict__ hop,    // [E,FEAT_DIM]
        const int* __restrict__ eidx,     // [2,E]
        const float* __restrict__ phase,  // [K,E,2] or nullptr
        const float* __restrict__ kpts,   // fallback phase inputs
        const float* __restrict__ shifts,
        float* __restrict__ out, int E, int K, int M) {
    __shared__ float lds_feat[EPB * FEAT_DIM];
    const int e0 = blockIdx.x * EPB;
    const int nrows = min(EPB, E - e0);

#if defined(HAVE_TDM)
    if (threadIdx.x < 32) {
        // Tensor DMA descriptor (D#), ISA 8.3-8.6. 2-D tile: 107 floats x nrows.
        uint64_t ga = (uint64_t)(uintptr_t)(hop + (size_t)e0 * FEAT_DIM);
        uint32_t lds_addr = (uint32_t)(uintptr_t)(void*)&lds_feat[0];  // low 32b = LDS offset
        u32x4 g0;
        g0[0] = 1u;                                            // count=1, user descriptor
        g0[1] = lds_addr;                                      // lds_addr
        g0[2] = (uint32_t)ga;                                  // global_addr[31:0]
        g0[3] = (uint32_t)((ga >> 32) & 0x1FFFFFFu) | (2u << 30);  // addr[56:32] | type=2
        i32x8 g1;
        g1[0] = 0x00020000;                  // wg_mask=0, data_size=2 (4B)
        g1[1] = (int)((FEAT_DIM & 0xFFFF) << 16);   // tensor_dim0[15:0] @ bits63:48
        g1[2] = (int)(((unsigned)E & 0xFFFF) << 16);// tensor_dim0 hi=0, tensor_dim1[15:0]
        g1[3] = (int)(((unsigned)FEAT_DIM) << 16);  // tensor_dim1 hi=0, tile_dim0=107
        g1[4] = nrows;                        // tile_dim1 = rows, tile_dim2 = 0
        g1[5] = FEAT_DIM;                     // tensor_dim0_stride[31:0]
        g1[6] = 0;                            // stride0 hi, stride1 lo
        g1[7] = 0;                            // stride1 hi
        i32x4 z4 = {0, 0, 0, 0};
#if defined(__clang_major__) && (__clang_major__ >= 23)
        i32x8 z8 = {0, 0, 0, 0, 0, 0, 0, 0};
        __builtin_amdgcn_tensor_load_to_lds(g0, g1, z4, z4, z8, 0);
#else
        __builtin_amdgcn_tensor_load_to_lds(g0, g1, z4, z4, 0);
#endif
        __builtin_amdgcn_s_wait_tensorcnt(0);
    }
#else
    for (int t = threadIdx.x; t < nrows * FEAT_DIM; t += blockDim.x)
        lds_feat[t] = hop[(size_t)e0 * FEAT_DIM + t];
#endif
    __syncthreads();

    const size_t mm = (size_t)M * M;
    for (int t = threadIdx.x; t < EPB * NORB * NORB; t += blockDim.x) {
        int le = t / (NORB * NORB);
        if (le >= nrows) continue;
        int pe = t - le * (NORB * NORB);
        int p = pe / NORB, q = pe - p * NORB;
        int i = orb_of(p), j = orb_of(q);
        if (i > j) continue;  // assembled block entry is zero
        float scale = (i == j) ? 0.5f : 1.0f;
        float h = scale * lds_feat[le * FEAT_DIM + feat_idx(i, j, p, q)];
        int e = e0 + le;
        int ai = eidx[e], aj = eidx[E + e];
        size_t row = (size_t)ai * NORB + p;
        size_t col = (size_t)aj * NORB + q;
        for (int k = 0; k < K; ++k) {
            float pr, pim;
            if (phase) {
                size_t pidx = 2 * ((size_t)k * E + e);
                pr = phase[pidx];
                pim = phase[pidx + 1];
            } else {
                float dot = kpts[k * 3 + 0] * shifts[e * 3 + 0] +
                            kpts[k * 3 + 1] * shifts[e * 3 + 1] +
                            kpts[k * 3 + 2] * shifts[e * 3 + 2];
                float s, cc;
                __sincosf(6.28318530717958647692f * dot, &s, &cc);
                pr = cc; pim = -s;
            }
            float re = pr * h, im = pim * h;
            float* o1 = out + 2 * ((size_t)k * mm + row * M + col);
            unsafeAtomicAdd(o1 + 0, re);
            unsafeAtomicAdd(o1 + 1, im);
            float* o2 = out + 2 * ((size_t)k * mm + col * M + row);  // conj term
            unsafeAtomicAdd(o2 + 0, re);
            unsafeAtomicAdd(o2 + 1, -im);
        }
    }
}

// ---------------------------------------------------------------------------
extern "C" void kernel_launch(void* const* d_in, const int* in_sizes, int n_in,
                              void* d_out, int out_size, void* d_ws, size_t ws_size,
                              hipStream_t stream) {
    const float* hop    = (const float*)d_in[0];  // [E,107]
    const float* ons    = (const float*)d_in[1];  // [N,107]
    const float* kpts   = (const float*)d_in[2];  // [K,3]
    const float* shifts = (const float*)d_in[3];  // [E,3]
    const int*   eidx   = (const int*)d_in[4];    // [2,E]
    float* out = (float*)d_out;                   // [K, M, M] complex interleaved

    const int E = in_sizes[0] / FEAT_DIM;
    const int N = in_sizes[1] / FEAT_DIM;
    const int K = in_sizes[2] / 3;
    const int M = N * NORB;

    // 1) zero the dense output (bandwidth-dominant step)
    zero_fill_kernel<<<8192, 256, 0, stream>>>(out, (long)out_size);

    // 2) phases via WMMA into workspace (fallback: inline in scatter kernel)
    float* phase = nullptr;
    size_t phase_bytes = (size_t)K * E * 2 * sizeof(float);
    if (ws_size >= phase_bytes && K <= 8) {
        phase = (float*)d_ws;
        int pblocks = (E + 15) / 16;
        phase_wmma_kernel<<<pblocks, 32, 0, stream>>>(kpts, shifts, phase, E, K);
    }

    // 3) onsite diagonal blocks (plain stores, stream-ordered after zero-fill)
    int ototal = N * NORB * NORB;
    onsite_kernel<<<(ototal + 255) / 256, 256, 0, stream>>>(ons, out, N, K, M);

    // 4) fused hermitian edge scatter (atomics, after onsite in stream order)
    int eblocks = (E + EPB - 1) / EPB;
    edge_scatter_kernel<<<eblocks, 256, 0, stream>>>(hop, eidx, phase, kpts,
                                                     shifts, out, E, K, M);
}